// HOPE_SSM_Kernel_62843961475574
// MI455X (gfx1250) — compile-verified
//
#include <hip/hip_runtime.h>
#include <math.h>

// HOPE SSM frequency-domain kernel generator for MI455X (gfx1250, wave32).
// K[h,l] = sum_n Hc[h,n] * exp(-i*ang(h,l)*(n+1)),  ang = angle(Mobius(e^{2pi i l/L}; dt_h))
//
// One wave handles (h, 16 consecutive l). The complex n-reduction is expressed
// as a real 16x16x64 GEMM executed as 16 chained V_WMMA_F32_16X16X4_F32 ops:
//   A[l, 2n]   = Re exp(-i*ang*(n+1))      A[l, 2n+1] = Im exp(-i*ang*(n+1))
//   B[2n, 0]   = H_r[n]   B[2n+1, 0] = -H_i[n]   (col 0 -> K_real)
//   B[2n, 1]   = H_i[n]   B[2n+1, 1] =  H_r[n]   (col 1 -> K_imag)
// Phase values come from a complex-rotation recurrence (2 sincos per lane total).

typedef __attribute__((ext_vector_type(2))) float v2f;
typedef __attribute__((ext_vector_type(8))) float v8f;

#define H_DIM   1024
#define N_HALF  32
#define SEQ_L   2048
#define WAVES_PER_BLOCK 8

__global__ __launch_bounds__(256) void hope_ssm_wmma_kernel(
    const float* __restrict__ log_dt,   // (h,)
    const float* __restrict__ H,        // (h, 32, 2) re/im interleaved
    float* __restrict__ out)            // (h, L, 2) complex64 interleaved
{
    __shared__ float sH[2 * N_HALF];

    const int h   = blockIdx.x;
    const int tid = threadIdx.x;

    if (tid < 2 * N_HALF) sH[tid] = H[h * 2 * N_HALF + tid];
    __syncthreads();

    const int wave  = tid >> 5;
    const int lane  = tid & 31;
    const int p     = lane & 15;        // l-row within tile / output column id
    const int hi    = lane >> 4;        // lane-half: owns K rows 4c+2hi+{0,1}
    const int lbase = (blockIdx.y * WAVES_PER_BLOCK + wave) * 16;
    const int l     = lbase + p;

    // ---- angle of the Mobius-warped unit-circle sample ----
    const float dt = __expf(log_dt[h]);
    const float theta = (float)l * (6.283185307179586f / (float)SEQ_L);
    float st, ct;
    __sincosf(theta, &st, &ct);
    const float nr = (1.0f + dt) * ct + (dt - 1.0f);   // num = (1+dt)s + (dt-1)
    const float ni = (1.0f + dt) * st;
    const float dr = (dt - 1.0f) * ct + (dt + 1.0f);   // den = (dt-1)s + (dt+1)
    const float di = (dt - 1.0f) * st;
    // angle(num/den) = atan2(Im(num*conj(den)), Re(num*conj(den)))
    const float ang = atan2f(ni * dr - nr * di, nr * dr + ni * di);

    // ---- phase recurrence: z_c = exp(-i*ang*(2c+1+hi)), step = exp(-i*2*ang)
    float s1, c1, s2, c2;
    __sincosf(ang * (float)(1 + hi), &s1, &c1);
    __sincosf(ang * 2.0f, &s2, &c2);
    float zr = c1, zi = -s1;
    const float er = c2, ei = -s2;

    v8f acc = {};
#pragma unroll
    for (int c = 0; c < 16; ++c) {
        const int n = 2 * c + hi;               // state index this lane-half feeds
        const float hr  = sH[2 * n];
        const float him = sH[2 * n + 1];

        v2f a;                                  // A: phase (Re, Im) for K rows
        a.x = zr;
        a.y = zi;

        v2f b;                                  // B: only columns 0 (real), 1 (imag)
        b.x = (p == 0) ? hr   : ((p == 1) ? him : 0.0f);
        b.y = (p == 0) ? -him : ((p == 1) ? hr  : 0.0f);

        acc = __builtin_amdgcn_wmma_f32_16x16x4_f32(
            /*neg_a=*/false, a, /*neg_b=*/false, b,
            /*c_mod=*/(short)0, acc, /*reuse_a=*/false, /*reuse_b=*/false);

        // z *= exp(-i*2*ang)
        const float tr = zr * er - zi * ei;
        zi = zr * ei + zi * er;
        zr = tr;
    }

    // ---- writeback: lane p=0 holds K_real column, p=1 holds K_imag column;
    //      D VGPR r maps to output row M = r + 8*hi.
    if (p < 2) {
        const int obase = (h * SEQ_L + lbase + 8 * hi) * 2 + p;
#pragma unroll
        for (int r = 0; r < 8; ++r)
            out[obase + 2 * r] = acc[r];
    }
}

extern "C" void kernel_launch(void* const* d_in, const int* in_sizes, int n_in,
                              void* d_out, int out_size, void* d_ws, size_t ws_size,
                              hipStream_t stream) {
    const float* log_dt = (const float*)d_in[0];   // (1024,)
    const float* H      = (const float*)d_in[1];   // (1024, 32, 2)
    // d_in[2] is L (scalar) -- compile-time constant SEQ_L here.
    float* out = (float*)d_out;                    // (1024, 2048, 2)

    dim3 grid(H_DIM, SEQ_L / (16 * WAVES_PER_BLOCK));  // (1024, 16)
    dim3 block(32 * WAVES_PER_BLOCK);                  // 256 threads = 8 waves
    hope_ssm_wmma_kernel<<<grid, block, 0, stream>>>(log_dt, H, out);
}